// FlattenScaledDotProduct_33509334843951
// MI455X (gfx1250) — compile-verified
//
#include <hip/hip_runtime.h>
#include <math.h>

// Problem constants (from the reference)
#define E_TOTAL   2000000
#define DIM       64
#define NSEG      100000
#define INV_TEMP  0.125f          // 1/8.0
#define NTILES    (E_TOTAL / 16)  // 125000, E divisible by 16

typedef float v2f __attribute__((ext_vector_type(2)));
typedef float v4f __attribute__((ext_vector_type(4)));
typedef float v8f __attribute__((ext_vector_type(8)));

// Explicit address-space types so LDS traffic lowers to DS instructions
// (not FLAT) and the async builtin gets its exact pointer types.
typedef __attribute__((address_space(3))) float LDSf;
typedef __attribute__((address_space(3))) v2f   LDSv2f;
typedef __attribute__((address_space(3))) v4f   LDSv4f;
typedef int v4i __attribute__((vector_size(16)));
typedef __attribute__((address_space(1))) v4i GV4;   // global int4
typedef __attribute__((address_space(3))) v4i LV4;   // LDS int4

// Padded LDS tile: 16 rows x 68 floats (pad keeps b128 staging and
// ds_load_b64 A-layout reads bank-conflict free: bank = 4r+4c+2h mod 64).
#define ROW_F   68
#define TILE_F  (16 * ROW_F)      // 1088 floats per matrix tile
#define WAVE_F  (2 * 2 * TILE_F)  // 2 buffers x {q,k}
#define WAVES_PER_BLOCK 8
#define BLOCK_LDS_BYTES (WAVES_PER_BLOCK * WAVE_F * 4)  // 139264 B

#if defined(__has_builtin)
#  if __has_builtin(__builtin_amdgcn_global_load_async_to_lds_b128) && \
      __has_builtin(__builtin_amdgcn_s_wait_asynccnt)
#    define USE_ASYNC 1
#  endif
#endif
#ifndef USE_ASYNC
#  define USE_ASYNC 0
#endif

#if USE_ASYNC
#  define WAIT_STAGE(n) __builtin_amdgcn_s_wait_asynccnt(n)
#else
#  define WAIT_STAGE(n) ((void)0)  // compiler tracks loadcnt/dscnt for fallback
#endif

// Stage one 16x64 f32 tile (4 KB) from global into padded LDS.
// 8 iterations x 32 lanes x b128: global side is fully lane-coalesced
// (consecutive lanes -> consecutive 16B), LDS side is conflict-free.
__device__ __forceinline__ void stage_tile(const float* __restrict__ g,
                                           LDSf* dst, long tile, int lane) {
  const float* base = g + (size_t)tile * (16 * DIM);
#pragma unroll
  for (int i = 0; i < 8; ++i) {
    int lin = i * 32 + lane;          // linear float4 id within tile
    int row = lin >> 4;               // edge row 0..15
    int col = (lin & 15) << 2;        // k-column 0..60
#if USE_ASYNC
    __builtin_amdgcn_global_load_async_to_lds_b128(
        (GV4*)(base + lin * 4),
        (LV4*)(dst + row * ROW_F + col),
        0, 0);
#else
    v4f v = *(const v4f*)(base + lin * 4);
    *(LDSv4f*)(dst + row * ROW_F + col) = v;
#endif
  }
}

// Kernel 1: per-edge scaled dot products via WMMA row-sum trick.
// A = elementwise (Q tile ∘ K tile) in the 16x4 f32 A layout
//     (lane%16 = row M; lanes<16 carry K=4c,4c+1; lanes>=16 carry K=4c+2,4c+3),
// B = all-ones  ->  D[m][n] = row sum of products = dot(q_m, k_m) in ALL columns.
// After 16 accumulating WMMAs over K=64: lanes 0..15 hold rows 0..7 in acc[0..7],
// lanes 16..31 hold rows 8..15.  Lanes 0 and 16 write the scores.
__global__ void __launch_bounds__(256)
k1_scores(const float* __restrict__ q, const float* __restrict__ k,
          float* __restrict__ sc) {
  extern __shared__ float smem[];
  LDSf* lds = (LDSf*)smem;            // one generic->AS3 cast, AS3 thereafter
  const int lane = threadIdx.x & 31;
  const int wv   = threadIdx.x >> 5;
  const int r    = lane & 15;   // edge row within tile
  const int half = lane >> 4;   // 0: K%4 in {0,1}, 1: K%4 in {2,3}

  LDSf* wbase = lds + wv * WAVE_F;    // [buf][q|k][TILE_F]

  const long nw  = (long)gridDim.x * WAVES_PER_BLOCK;
  long wid = (long)blockIdx.x * WAVES_PER_BLOCK + wv;
  if (wid >= NTILES) return;

  stage_tile(q, wbase, wid, lane);
  stage_tile(k, wbase + TILE_F, wid, lane);

  const v2f ones = {1.0f, 1.0f};
  int cur = 0;
  for (long t = wid; t < NTILES; t += nw) {
    long tn = t + nw;
    LDSf* curb = wbase + cur * (2 * TILE_F);
    LDSf* nxtb = wbase + (cur ^ 1) * (2 * TILE_F);
    if (tn < NTILES) {                       // double-buffered prefetch
      stage_tile(q, nxtb, tn, lane);
      stage_tile(k, nxtb + TILE_F, tn, lane);
      WAIT_STAGE(16);                        // async ops complete in order:
    } else {                                 // first 16 (current tile) done
      WAIT_STAGE(0);
    }

    const LDSf* qrow = curb + r * ROW_F + half * 2;
    const LDSf* krow = qrow + TILE_F;
    v8f acc = {0.f, 0.f, 0.f, 0.f, 0.f, 0.f, 0.f, 0.f};
#pragma unroll
    for (int c = 0; c < 16; ++c) {
      v2f qa = *(const LDSv2f*)(qrow + c * 4);  // ds_load_b64, conflict-free
      v2f ka = *(const LDSv2f*)(krow + c * 4);
      v2f p  = qa * ka;                         // A chunk: P[r][4c+2h .. +1]
      acc = __builtin_amdgcn_wmma_f32_16x16x4_f32(
          false, p, false, ones, (short)0, acc, false, false);
    }

    if (r == 0) {                            // lanes 0 and 16
      size_t eb = (size_t)t * 16 + (size_t)half * 8;
      sc[eb + 0] = acc[0] * INV_TEMP;
      sc[eb + 1] = acc[1] * INV_TEMP;
      sc[eb + 2] = acc[2] * INV_TEMP;
      sc[eb + 3] = acc[3] * INV_TEMP;
      sc[eb + 4] = acc[4] * INV_TEMP;
      sc[eb + 5] = acc[5] * INV_TEMP;
      sc[eb + 6] = acc[6] * INV_TEMP;
      sc[eb + 7] = acc[7] * INV_TEMP;
    }
    cur ^= 1;
  }
}

// Kernel 2: one thread per segment. index is sorted, so each segment is a
// contiguous range found by binary search. Deterministic (no atomics).
__global__ void seg_reduce(const float* __restrict__ sc,
                           const int* __restrict__ idx,
                           float* __restrict__ segmax,
                           float* __restrict__ denom) {
  int n = blockIdx.x * blockDim.x + threadIdx.x;
  if (n >= NSEG) return;
  // lower_bound for n, then for n+1
  int lo = 0, hi = E_TOTAL;
  while (lo < hi) { int m = (lo + hi) >> 1; if (idx[m] <  n) lo = m + 1; else hi = m; }
  int s0 = lo;
  hi = E_TOTAL;
  while (lo < hi) { int m = (lo + hi) >> 1; if (idx[m] <= n) lo = m + 1; else hi = m; }
  int s1 = lo;

  float mx = -INFINITY;
  for (int e = s0; e < s1; ++e) mx = fmaxf(mx, sc[e]);
  float d = 0.0f;
  for (int e = s0; e < s1; ++e) d += expf(sc[e] - mx);
  segmax[n] = mx;
  denom[n]  = d;
}

// Kernel 3: per-edge normalize.
__global__ void finalize(const float* __restrict__ sc,
                         const int* __restrict__ idx,
                         const float* __restrict__ segmax,
                         const float* __restrict__ denom,
                         float* __restrict__ out) {
  int e = blockIdx.x * blockDim.x + threadIdx.x;
  if (e >= E_TOTAL) return;
  int n = idx[e];
  out[e] = expf(sc[e] - segmax[n]) / denom[n];
}

extern "C" void kernel_launch(void* const* d_in, const int* in_sizes, int n_in,
                              void* d_out, int out_size, void* d_ws, size_t ws_size,
                              hipStream_t stream) {
  (void)in_sizes; (void)n_in; (void)out_size; (void)ws_size;
  const float* q  = (const float*)d_in[0];
  const float* k  = (const float*)d_in[1];
  const int*  idx = (const int*)d_in[2];
  float* out = (float*)d_out;

  // workspace layout: scores [E] | segmax [NSEG] | denom [NSEG]  (~8.8 MB)
  float* sc     = (float*)d_ws;
  float* segmax = sc + E_TOTAL;
  float* denom  = segmax + NSEG;

  k1_scores<<<1024, 256, BLOCK_LDS_BYTES, stream>>>(q, k, sc);
  seg_reduce<<<(NSEG + 255) / 256, 256, 0, stream>>>(sc, idx, segmax, denom);
  finalize<<<(E_TOTAL + 255) / 256, 256, 0, stream>>>(sc, idx, segmax, denom, out);
}